// BasicModule_43911745634888
// MI455X (gfx1250) — compile-verified
//
#include <hip/hip_runtime.h>
#include <math.h>

typedef __attribute__((ext_vector_type(2))) float v2f;
typedef __attribute__((ext_vector_type(4))) float v4f;
typedef __attribute__((ext_vector_type(8))) float v8f;

#define S_DIM 2048
#define B_DIM 64
#define H_DIM 1024
#define NCHUNK 16
#define CHUNK (S_DIM / NCHUNK) /* 128 */
#define BN_EPS 1e-5f

// ---------------------------------------------------------------------------
// Pass 1: single pass over ctx (512 MB). One block per (b, s-chunk).
// Each thread owns 4 consecutive h values (tid*4 .. tid*4+3).
// Online softmax: running max m, running sum l, exp-weighted acc in registers.
// ---------------------------------------------------------------------------
__global__ __launch_bounds__(256) void attn_chunk_kernel(
    const float* __restrict__ ctx, const float* __restrict__ key,
    const float* __restrict__ mask, float* __restrict__ scores,
    float* __restrict__ accs, float* __restrict__ mvals,
    float* __restrict__ lvals)
{
    __shared__ float sred[8];
    const int b    = blockIdx.x >> 4;
    const int sc   = blockIdx.x & 15;
    const int tid  = threadIdx.x;
    const int lane = tid & 31;
    const int wave = tid >> 5;

    // This thread's 4 key values (reused every iteration, stays in registers)
    const v4f kreg = *(const v4f*)(key + b * H_DIM + tid * 4);

    float m = -3.402823466e+38f;
    float l = 0.0f;
    v4f acc = {0.0f, 0.0f, 0.0f, 0.0f};

    const float* base = ctx + ((size_t)sc * CHUNK * B_DIM + b) * H_DIM;
    for (int si = 0; si < CHUNK; ++si) {
        const float* crow = base + (size_t)si * (B_DIM * H_DIM);
        const v4f v = *(const v4f*)(crow + tid * 4);
        if (si + 1 < CHUNK) // prefetch next s-row (256 KB stride)
            __builtin_prefetch(crow + (size_t)B_DIM * H_DIM + tid * 4, 0, 0);

        // per-thread partial dot, then wave32 shuffle reduce
        float part = v.x * kreg.x + v.y * kreg.y + v.z * kreg.z + v.w * kreg.w;
        #pragma unroll
        for (int off = 16; off > 0; off >>= 1)
            part += __shfl_xor(part, off, 32);
        if (lane == 0) sred[wave] = part;
        __syncthreads();
        const float dot = sred[0] + sred[1] + sred[2] + sred[3] +
                          sred[4] + sred[5] + sred[6] + sred[7];

        const int s = sc * CHUNK + si;
        const float score = dot + mask[b * S_DIM + s];
        if (tid == 0) scores[b * S_DIM + s] = score;

        // online softmax update
        const float mnew  = fmaxf(m, score);
        const float scale = __expf(m - mnew);      // 0 on first iter
        const float w     = __expf(score - mnew);
        l = l * scale + w;
        acc.x = acc.x * scale + w * v.x;
        acc.y = acc.y * scale + w * v.y;
        acc.z = acc.z * scale + w * v.z;
        acc.w = acc.w * scale + w * v.w;
        m = mnew;
        __syncthreads();   // protect sred before next iteration
    }

    float* accp = accs + ((size_t)b * NCHUNK + sc) * H_DIM;
    *(v4f*)(accp + tid * 4) = acc;
    if (tid == 0) {
        mvals[b * NCHUNK + sc] = m;
        lvals[b * NCHUNK + sc] = l;
    }
}

// ---------------------------------------------------------------------------
// Pass 2: combine the 16 chunk accumulators per batch; build x0 = [attn, key].
// ---------------------------------------------------------------------------
__global__ __launch_bounds__(256) void attn_combine_kernel(
    const float* __restrict__ accs, const float* __restrict__ mvals,
    const float* __restrict__ lvals, const float* __restrict__ key,
    float* __restrict__ x0, float* __restrict__ Mb, float* __restrict__ Lb)
{
    __shared__ float sw[NCHUNK];
    __shared__ float sinvL;
    const int b = blockIdx.x;
    const int tid = threadIdx.x;

    if (tid == 0) {
        float M = -3.402823466e+38f;
        for (int c = 0; c < NCHUNK; ++c) M = fmaxf(M, mvals[b * NCHUNK + c]);
        float L = 0.0f;
        for (int c = 0; c < NCHUNK; ++c) {
            const float w = __expf(mvals[b * NCHUNK + c] - M);
            sw[c] = w;
            L += lvals[b * NCHUNK + c] * w;
        }
        Mb[b] = M;
        Lb[b] = L;
        sinvL = 1.0f / L;
    }
    __syncthreads();

    v4f a = {0.0f, 0.0f, 0.0f, 0.0f};
    for (int c = 0; c < NCHUNK; ++c) {
        const v4f p = *(const v4f*)(accs + ((size_t)b * NCHUNK + c) * H_DIM + tid * 4);
        const float w = sw[c];
        a.x += w * p.x; a.y += w * p.y; a.z += w * p.z; a.w += w * p.w;
    }
    const float inv = sinvL;
    a.x *= inv; a.y *= inv; a.z *= inv; a.w *= inv;

    *(v4f*)(x0 + (size_t)b * 2 * H_DIM + tid * 4) = a;
    *(v4f*)(x0 + (size_t)b * 2 * H_DIM + H_DIM + tid * 4) =
        *(const v4f*)(key + b * H_DIM + tid * 4);
}

// ---------------------------------------------------------------------------
// Pass 3: At[b][s] = exp(score - M_b) / L_b  (second tuple output)
// ---------------------------------------------------------------------------
__global__ __launch_bounds__(256) void at_kernel(
    const float* __restrict__ scores, const float* __restrict__ Mb,
    const float* __restrict__ Lb, float* __restrict__ At)
{
    const int idx = blockIdx.x * 256 + threadIdx.x;
    const int b = idx >> 11; // / S_DIM
    At[idx] = __expf(scores[idx] - Mb[b]) / Lb[b];
}

// ---------------------------------------------------------------------------
// fp32 WMMA GEMM: out[m][n] = sum_k A[m][k] * W[n][k] + bias[n]
// One wave per 16x16 output tile, K-loop in steps of 4 via
// V_WMMA_F32_16X16X4_F32.  A/B operands: lane L<16 holds row/col L, K={0,1};
// lanes 16..31 hold K={2,3}  ->  single aligned float2 load each.
// C/D layout: VGPR v <-> row v + 8*hi, lane%16 <-> col.
// ---------------------------------------------------------------------------
__global__ __launch_bounds__(256) void gemm_wmma_kernel(
    const float* __restrict__ A, const float* __restrict__ W,
    const float* __restrict__ bias, float* __restrict__ out,
    int N, int K)
{
    const int waveId = blockIdx.x * 8 + (threadIdx.x >> 5);
    const int lane   = threadIdx.x & 31;
    const int tilesN = N >> 4;
    const int m0 = (waveId / tilesN) << 4;
    const int n0 = (waveId % tilesN) << 4;
    const int r  = lane & 15;
    const int hi = lane >> 4;

    const float* arow = A + (size_t)(m0 + r) * K + 2 * hi;
    const float* wrow = W + (size_t)(n0 + r) * K + 2 * hi;

    v8f c = {0.0f, 0.0f, 0.0f, 0.0f, 0.0f, 0.0f, 0.0f, 0.0f};
    #pragma unroll 4
    for (int k = 0; k < K; k += 4) {
        const v2f av = *(const v2f*)(arow + k);
        const v2f bv = *(const v2f*)(wrow + k);
        c = __builtin_amdgcn_wmma_f32_16x16x4_f32(
                /*neg_a=*/false, av, /*neg_b=*/false, bv,
                /*c_mod=*/(short)0, c, /*reuse_a=*/false, /*reuse_b=*/false);
    }

    const float bb = bias[n0 + r];
    #pragma unroll
    for (int v = 0; v < 8; ++v)
        out[(size_t)(m0 + v + 8 * hi) * N + n0 + r] = c[v] + bb;
}

// ---------------------------------------------------------------------------
// BatchNorm1d (training stats, biased var) + tanh.  One thread per channel.
// ---------------------------------------------------------------------------
__global__ __launch_bounds__(256) void bn_tanh_kernel(
    const float* __restrict__ x1, const float* __restrict__ gamma,
    const float* __restrict__ beta, float* __restrict__ y)
{
    const int c = blockIdx.x * 256 + threadIdx.x;
    float sum = 0.0f, sumsq = 0.0f;
    for (int i = 0; i < B_DIM; ++i) {
        const float v = x1[i * H_DIM + c];
        sum += v;
        sumsq += v * v;
    }
    const float mean = sum * (1.0f / B_DIM);
    const float var  = sumsq * (1.0f / B_DIM) - mean * mean;
    const float sc   = rsqrtf(var + BN_EPS) * gamma[c];
    const float sh   = beta[c] - mean * sc;
    for (int i = 0; i < B_DIM; ++i)
        y[i * H_DIM + c] = tanhf(x1[i * H_DIM + c] * sc + sh);
}

// ---------------------------------------------------------------------------
extern "C" void kernel_launch(void* const* d_in, const int* in_sizes, int n_in,
                              void* d_out, int out_size, void* d_ws, size_t ws_size,
                              hipStream_t stream) {
    const float* ctx   = (const float*)d_in[0];
    const float* key   = (const float*)d_in[1];
    const float* mask  = (const float*)d_in[2];
    const float* fc1_w = (const float*)d_in[3];
    const float* fc1_b = (const float*)d_in[4];
    const float* gamma = (const float*)d_in[5];
    const float* beta  = (const float*)d_in[6];
    const float* fc2_w = (const float*)d_in[7];
    const float* fc2_b = (const float*)d_in[8];

    float* out_x  = (float*)d_out;                       // (B, H)
    float* out_At = (float*)d_out + B_DIM * H_DIM;       // (B, S)

    float* ws     = (float*)d_ws;
    float* scores = ws;                                  // B*S      = 131072
    float* accs   = scores + B_DIM * S_DIM;              // B*16*H   = 1048576
    float* mvals  = accs + (size_t)B_DIM * NCHUNK * H_DIM; // 1024
    float* lvals  = mvals + B_DIM * NCHUNK;              // 1024
    float* Mb     = lvals + B_DIM * NCHUNK;              // 64
    float* Lb     = Mb + B_DIM;                          // 64
    float* x0     = Lb + B_DIM;                          // B*2H     = 131072
    float* x1     = x0 + B_DIM * 2 * H_DIM;              // B*H      = 65536
    float* y      = x1 + B_DIM * H_DIM;                  // B*H      = 65536

    // 1) fused scores + online-softmax weighted sum (single pass over ctx)
    attn_chunk_kernel<<<B_DIM * NCHUNK, 256, 0, stream>>>(
        ctx, key, mask, scores, accs, mvals, lvals);

    // 2) combine chunks -> attn, build x0 = [attn, key]
    attn_combine_kernel<<<B_DIM, 256, 0, stream>>>(
        accs, mvals, lvals, key, x0, Mb, Lb);

    // 3) At output
    at_kernel<<<(B_DIM * S_DIM) / 256, 256, 0, stream>>>(scores, Mb, Lb, out_At);

    // 4) x1 = x0 @ fc1_w^T + fc1_b   (fp32 WMMA), tiles = 4*64 = 256 waves
    gemm_wmma_kernel<<<(4 * (H_DIM / 16)) / 8, 256, 0, stream>>>(
        x0, fc1_w, fc1_b, x1, H_DIM, 2 * H_DIM);

    // 5) batchnorm (training stats) + tanh
    bn_tanh_kernel<<<H_DIM / 256, 256, 0, stream>>>(x1, gamma, beta, y);

    // 6) out = y @ fc2_w^T + fc2_b   (fp32 WMMA)
    gemm_wmma_kernel<<<(4 * (H_DIM / 16)) / 8, 256, 0, stream>>>(
        y, fc2_w, fc2_b, out_x, H_DIM, H_DIM);
}